// LengthRegulator_9603546874190
// MI455X (gfx1250) — compile-verified
//
#include <hip/hip_runtime.h>
#include <stdint.h>

// Problem constants (from reference): B=16, T=1024, D=384, MAX_DUR=4.
#define B_  16
#define T_  1024
#define D_  384
#define OUT_LEN (T_ * 4)            // 4096 padded frames per batch
#define ROW_BYTES (D_ * 4)          // 1536 bytes per frame
#define WAVES_PER_BLOCK 8           // 256 threads, wave32
#define ROWS_PER_WAVE 4
#define ROWS_PER_BLOCK (WAVES_PER_BLOCK * ROWS_PER_WAVE)  // 32
#define ELEMS_PER_THREAD (T_ / (WAVES_PER_BLOCK * 32))    // 4 durations/thread

// ---------------------------------------------------------------------------
// Single fused kernel: per-block recompute of the batch cumsum (cheap: one
// int4 load + 5-step __shfl_up wave scan + 8-entry LDS exchange, 2 barriers),
// then gather output rows through the CDNA5 async LDS data mover (ASYNCcnt):
//   global_load_async_to_lds_b128  (3x, 512B each = full 1536B row)
//   s_wait_asynccnt 0
//   global_store_async_from_lds_b128 (3x)
// Double-buffered so row r+1 loads overlap row r stores. Pad rows are plain
// b128 zero stores. All branches are wave-uniform (row index is per-wave).
// ---------------------------------------------------------------------------
__global__ __launch_bounds__(WAVES_PER_BLOCK * 32) void lr_fused(
    const float* __restrict__ xs, const int* __restrict__ ds,
    float* __restrict__ out) {
  __shared__ int scs[T_];                  // inclusive cumsum of durations
  __shared__ int wsum[WAVES_PER_BLOCK];    // per-wave duration sums
  __shared__ __align__(16) unsigned char stage[WAVES_PER_BLOCK][2][ROW_BYTES];

  const int b = blockIdx.y;
  const int tid = threadIdx.x;
  const int wave = tid >> 5;
  const int lane = tid & 31;

  // ---- in-block inclusive scan of ds[b, 0:1024] ----
  const int4 dv = ((const int4*)(ds + (size_t)b * T_))[tid];  // 4 durations
  const int s0 = dv.x;
  const int s1 = s0 + dv.y;
  const int s2 = s1 + dv.z;
  const int s3 = s2 + dv.w;

  int incl = s3;  // wave-level inclusive scan of per-thread sums
  #pragma unroll
  for (int off = 1; off < 32; off <<= 1) {
    int n = __shfl_up(incl, off, 32);
    if (lane >= off) incl += n;
  }
  if (lane == 31) wsum[wave] = incl;
  __syncthreads();

  int waveBase = 0, total = 0;
  #pragma unroll
  for (int w = 0; w < WAVES_PER_BLOCK; ++w) {
    int v = wsum[w];
    waveBase += (w < wave) ? v : 0;
    total += v;
  }
  const int tBase0 = waveBase + incl - s3;  // exclusive prefix of this thread
  scs[tid * ELEMS_PER_THREAD + 0] = tBase0 + s0;
  scs[tid * ELEMS_PER_THREAD + 1] = tBase0 + s1;
  scs[tid * ELEMS_PER_THREAD + 2] = tBase0 + s2;
  scs[tid * ELEMS_PER_THREAD + 3] = tBase0 + s3;
  __syncthreads();

  // ---- gather: one wave per output frame row ----
  // LDS byte offsets of this wave's two stage buffers (per-lane 16B slot);
  // generic pointer to LDS carries the group-relative offset in low 32 bits.
  const unsigned ldsA = (unsigned)(size_t)(&stage[wave][0][0]) + (unsigned)(lane * 16);
  const unsigned ldsB = (unsigned)(size_t)(&stage[wave][1][0]) + (unsigned)(lane * 16);

  const int tRowBase = blockIdx.x * ROWS_PER_BLOCK + wave * ROWS_PER_WAVE;
  int parity = 0;

  for (int r = 0; r < ROWS_PER_WAVE; ++r) {
    const int t = tRowBase + r;
    float* dstRow = out + ((size_t)b * OUT_LEN + (size_t)t) * D_;

    if (t < total) {
      // searchsorted(cs, t, side='right'): first i with cs[i] > t
      int lo = 0, hi = T_;
      while (lo < hi) {
        int mid = (lo + hi) >> 1;
        if (scs[mid] <= t) lo = mid + 1; else hi = mid;
      }
      const int idx = (lo < T_) ? lo : (T_ - 1);

      const uint64_t src =
          (uint64_t)(const char*)(xs + ((size_t)b * T_ + (size_t)idx) * D_) +
          (uint64_t)(lane * 16);
      const uint64_t dst = (uint64_t)(char*)dstRow + (uint64_t)(lane * 16);
      const unsigned lds = parity ? ldsB : ldsA;

      // 3 x 512B async copies; IOFFSET applies to both global and LDS sides.
      asm volatile(
          "global_load_async_to_lds_b128 %0, %1, off\n\t"
          "global_load_async_to_lds_b128 %0, %1, off offset:512\n\t"
          "global_load_async_to_lds_b128 %0, %1, off offset:1024"
          :: "v"(lds), "v"(src) : "memory");
      // Retire these loads (and the previous row's stores — closes the
      // double-buffer reuse hazard two valid rows later).
      asm volatile("s_wait_asynccnt 0x0" ::: "memory");
      asm volatile(
          "global_store_async_from_lds_b128 %0, %1, off\n\t"
          "global_store_async_from_lds_b128 %0, %1, off offset:512\n\t"
          "global_store_async_from_lds_b128 %0, %1, off offset:1024"
          :: "v"(dst), "v"(lds) : "memory");
      parity ^= 1;
    } else {
      // Pad region: write zeros directly (96 float4 per row, 3 per lane).
      float4 z = make_float4(0.f, 0.f, 0.f, 0.f);
      float4* d4 = (float4*)dstRow;
      d4[lane] = z;
      d4[lane + 32] = z;
      d4[lane + 64] = z;
    }
  }
  // Drain outstanding async stores before wave exit.
  asm volatile("s_wait_asynccnt 0x0" ::: "memory");
}

// ---------------------------------------------------------------------------
extern "C" void kernel_launch(void* const* d_in, const int* in_sizes, int n_in,
                              void* d_out, int out_size, void* d_ws, size_t ws_size,
                              hipStream_t stream) {
  (void)in_sizes; (void)n_in; (void)out_size; (void)d_ws; (void)ws_size;
  const float* xs = (const float*)d_in[0];
  const int* ds = (const int*)d_in[1];
  float* out = (float*)d_out;

  lr_fused<<<dim3(OUT_LEN / ROWS_PER_BLOCK, B_), dim3(WAVES_PER_BLOCK * 32), 0,
             stream>>>(xs, ds, out);
}